// Matcher_87170656239838
// MI455X (gfx1250) — compile-verified
//
#include <hip/hip_runtime.h>
#include <math.h>

// Problem constants (fixed by the reference harness).
#define BS   8
#define NQ   16384
#define NG   1024
#define NC   5
#define TOPK 4
#define WAVES_PER_BLOCK 8
#define QTILES (NQ / 16)          // 1024 tiles of 16 queries

typedef __attribute__((ext_vector_type(2))) float v2f;
typedef __attribute__((ext_vector_type(8))) float v8f;

// ---------------------------------------------------------------------------
// Kernel 1: row softmax over NC=5 classes. 131072 rows, one thread per row.
// ---------------------------------------------------------------------------
__global__ __launch_bounds__(256) void softmax5_kernel(
    const float* __restrict__ logits, float* __restrict__ probs, int nrows) {
  int row = blockIdx.x * blockDim.x + threadIdx.x;
  if (row >= nrows) return;
  const float* l = logits + (size_t)row * NC;
  float l0 = l[0], l1 = l[1], l2 = l[2], l3 = l[3], l4 = l[4];
  float m = fmaxf(fmaxf(fmaxf(l0, l1), fmaxf(l2, l3)), l4);
  float e0 = __expf(l0 - m), e1 = __expf(l1 - m), e2 = __expf(l2 - m),
        e3 = __expf(l3 - m), e4 = __expf(l4 - m);
  float inv = 1.0f / (e0 + e1 + e2 + e3 + e4);
  float* p = probs + (size_t)row * NC;
  p[0] = e0 * inv; p[1] = e1 * inv; p[2] = e2 * inv; p[3] = e3 * inv; p[4] = e4 * inv;
}

// Branchless insert of candidate (c, qi) into sorted-ascending (tv0..tv3 / ti0..ti3).
// Strict '<' keeps earlier-inserted (smaller q) entries on ties -> matches
// jax.lax.top_k stability.
#define INSERT4(c, qi)                                                     \
  do {                                                                     \
    bool l3 = (c) < tv3, l2 = (c) < tv2, l1 = (c) < tv1, l0 = (c) < tv0;   \
    tv3 = l2 ? tv2 : (l3 ? (c) : tv3); ti3 = l2 ? ti2 : (l3 ? (qi) : ti3); \
    tv2 = l1 ? tv1 : (l2 ? (c) : tv2); ti2 = l1 ? ti1 : (l2 ? (qi) : ti2); \
    tv1 = l0 ? tv0 : (l1 ? (c) : tv1); ti1 = l0 ? ti0 : (l1 ? (qi) : ti1); \
    tv0 = l0 ? (c) : tv0;              ti0 = l0 ? (qi) : ti0;              \
  } while (0)

// ---------------------------------------------------------------------------
// Kernel 2: fused cost + top-4. One block per (batch, 16-gt column tile).
// Per 16x16 tile, TWO WMMAs:
//   WMMA1: A1_q = [1, |p|^2, x, y], B1_g = 0.01*[|g|^2, 1, -2gx, -2gy]
//          -> D1 = (0.1*dist)^2 tile (0.1 weight pre-folded, so sqrt gives
//             the weighted point cost directly).
//   VALU:  s = v_sqrt_f32(med3(0, D1, +inf))   (single-op clamp; fast HW
//          sqrt -- only per-column ordering matters for the index output)
//   WMMA2: A2_q = [p0,p1,p2,p3], B2_g = (lab==4 ? [1,1,1,1] : -onehot(lab)),
//          C = s  -> D2 = cost tile (+1 per column when lab==4; a per-column
//          constant offset cannot change that column's top-k indices).
// Accumulator layout (wave32): lane = half*16 + n holds rows M = half*8 + r
// of column N = n. Column g lives in lanes (g%16) and (g%16)+16; merged with
// __shfl_xor(16), then across the 8 waves via LDS.
// ---------------------------------------------------------------------------
__global__ __launch_bounds__(256) void matcher_topk_kernel(
    const float* __restrict__ pred_coords,        // [BS,NQ,2]
    const float* __restrict__ probs,              // [BS,NQ,NC] (from kernel 1)
    const float* __restrict__ gt_coords,          // [BS,NG,2]
    const int*   __restrict__ gt_labels,          // [BS,NG] (int32 on device)
    const unsigned char* __restrict__ gt_masks,   // [BS,NG] (bool8)
    int* __restrict__ out)                        // [BS,TOPK,NG]
{
  __shared__ float s_val[WAVES_PER_BLOCK][16][TOPK];
  __shared__ int   s_idx[WAVES_PER_BLOCK][16][TOPK];

  const int b     = blockIdx.x / (NG / 16);
  const int gbase = (blockIdx.x % (NG / 16)) * 16;

  const int tid  = threadIdx.x;
  const int wave = tid >> 5;
  const int lane = tid & 31;
  const int half = lane >> 4;   // 0: rows 0..7, 1: rows 8..15 of each tile
  const int lid  = lane & 15;   // column within the 16-gt tile
  const int g    = gbase + lid;

  // ---- per-column invariants ----
  const float gx = gt_coords[((size_t)b * NG + g) * 2 + 0];
  const float gy = gt_coords[((size_t)b * NG + g) * 2 + 1];
  const float b2 = gx * gx + gy * gy;
  const int  lab = gt_labels[(size_t)b * NG + g];

  // B1 (4x16), scaled by 0.01 = COST_POINT^2. VGPR0 holds K=(half?2:0),
  // VGPR1 holds K=(half?3:1), N=lid.
  v2f bf1;
  bf1.x = half ? (-0.02f * gx) : (0.01f * b2);
  bf1.y = half ? (-0.02f * gy) : 0.01f;

  // B2 (4x16): class-cost selector (see header comment).
  const int k0 = half ? 2 : 0, k1 = half ? 3 : 1;
  v2f bf2;
  bf2.x = (lab == 4) ? 1.0f : ((lab == k0) ? -1.0f : 0.0f);
  bf2.y = (lab == 4) ? 1.0f : ((lab == k1) ? -1.0f : 0.0f);

  const float* pc = pred_coords + (size_t)b * NQ * 2;
  const float* pp = probs       + (size_t)b * NQ * NC;

  const float INF = __builtin_inff();
  float tv0 = INF, tv1 = INF, tv2 = INF, tv3 = INF;
  int   ti0 = 0,   ti1 = 0,   ti2 = 0,   ti3 = 0;

  for (int t = wave; t < QTILES; t += WAVES_PER_BLOCK) {
    const int qbase = t * 16;
    const int qa    = qbase + lid;          // this lane's A-fragment row

    // A1: row M=lid. VGPR0 K=(half?2:0), VGPR1 K=(half?3:1).
    const float qx = pc[(size_t)qa * 2 + 0];
    const float qy = pc[(size_t)qa * 2 + 1];
    v2f af1;
    af1.x = half ? qx : 1.0f;
    af1.y = half ? qy : (qx * qx + qy * qy);

    // A2: probs p0..p3 of row M=lid; half 0 takes (p0,p1), half 1 (p2,p3).
    const float* pq = pp + (size_t)qa * NC + half * 2;
    v2f af2;
    af2.x = pq[0];
    af2.y = pq[1];

    v8f cz = {};
    v8f d2 = __builtin_amdgcn_wmma_f32_16x16x4_f32(
        false, af1, false, bf1, (short)0, cz, false, false);

    v8f s;
#pragma unroll
    for (int r = 0; r < 8; ++r) {
      // clamp to [0, +inf) in one v_med3_f32 (d2 is finite, never sNaN),
      // then raw v_sqrt_f32.
      float cl = __builtin_amdgcn_fmed3f(0.0f, d2[r], INF);
      s[r] = __builtin_amdgcn_sqrtf(cl);
    }

    v8f cost = __builtin_amdgcn_wmma_f32_16x16x4_f32(
        false, af2, false, bf2, (short)0, s, false, false);

    const int qrow0 = qbase + half * 8;
#pragma unroll
    for (int r = 0; r < 8; ++r) {
      const float c = cost[r];
      const int   q = qrow0 + r;
      INSERT4(c, q);
    }
  }

  // ---- merge the two lane-halves of each column (rows 0..7 vs 8..15) ----
  float pv0 = __shfl_xor(tv0, 16), pv1 = __shfl_xor(tv1, 16),
        pv2 = __shfl_xor(tv2, 16), pv3 = __shfl_xor(tv3, 16);
  int   pi0 = __shfl_xor(ti0, 16), pi1 = __shfl_xor(ti1, 16),
        pi2 = __shfl_xor(ti2, 16), pi3 = __shfl_xor(ti3, 16);
  INSERT4(pv0, pi0); INSERT4(pv1, pi1); INSERT4(pv2, pi2); INSERT4(pv3, pi3);

  if (half == 0) {
    s_val[wave][lid][0] = tv0; s_val[wave][lid][1] = tv1;
    s_val[wave][lid][2] = tv2; s_val[wave][lid][3] = tv3;
    s_idx[wave][lid][0] = ti0; s_idx[wave][lid][1] = ti1;
    s_idx[wave][lid][2] = ti2; s_idx[wave][lid][3] = ti3;
  }
  __syncthreads();

  // ---- final cross-wave merge + output (first 16 threads) ----
  if (tid < 16) {
    tv0 = INF; tv1 = INF; tv2 = INF; tv3 = INF;
    ti0 = 0;   ti1 = 0;   ti2 = 0;   ti3 = 0;
#pragma unroll
    for (int w = 0; w < WAVES_PER_BLOCK; ++w) {
#pragma unroll
      for (int j = 0; j < TOPK; ++j) {
        float c = s_val[w][tid][j];
        int   q = s_idx[w][tid][j];
        INSERT4(c, q);
      }
    }
    const int gg = gbase + tid;
    // Masked column: C = +inf for all q; top_k of equal values returns 0,1,2,3.
    const bool live = gt_masks[(size_t)b * NG + gg] != 0;
    out[((size_t)b * TOPK + 0) * NG + gg] = live ? ti0 : 0;
    out[((size_t)b * TOPK + 1) * NG + gg] = live ? ti1 : 1;
    out[((size_t)b * TOPK + 2) * NG + gg] = live ? ti2 : 2;
    out[((size_t)b * TOPK + 3) * NG + gg] = live ? ti3 : 3;
  }
}

extern "C" void kernel_launch(void* const* d_in, const int* in_sizes, int n_in,
                              void* d_out, int out_size, void* d_ws, size_t ws_size,
                              hipStream_t stream) {
  const float*         pred_coords = (const float*)d_in[0];
  const float*         pred_logits = (const float*)d_in[1];
  const float*         gt_coords   = (const float*)d_in[2];
  const int*           gt_labels   = (const int*)d_in[3];
  const unsigned char* gt_masks    = (const unsigned char*)d_in[4];
  int*   out   = (int*)d_out;
  float* probs = (float*)d_ws;   // BS*NQ*NC floats = 2.62 MB of workspace

  const int nrows = BS * NQ;
  softmax5_kernel<<<(nrows + 255) / 256, 256, 0, stream>>>(pred_logits, probs, nrows);
  matcher_topk_kernel<<<BS * (NG / 16), 256, 0, stream>>>(
      pred_coords, probs, gt_coords, gt_labels, gt_masks, out);
}